// STGCN_1202590843223
// MI455X (gfx1250) — compile-verified
//
#include <hip/hip_runtime.h>
#include <hip/hip_bf16.h>

typedef float v2f __attribute__((ext_vector_type(2)));
typedef float v8f __attribute__((ext_vector_type(8)));

// ---------------------------------------------------------------------------
// Degree / normalization kernels
// ---------------------------------------------------------------------------
__global__ void init_deg_kernel(float* deg, int n) {
    int i = blockIdx.x * blockDim.x + threadIdx.x;
    if (i < n) deg[i] = 1.0f;  // self-loop contributes 1
}

__global__ void accum_deg_kernel(const long long* __restrict__ ei, float* deg, int E) {
    int e = blockIdx.x * blockDim.x + threadIdx.x;
    if (e < E) {
        int c = (int)ei[(size_t)E + e];  // target of edge
        atomicAdd(&deg[c], 1.0f);
    }
}

__global__ void dinv_kernel(float* deg, int n) {
    int i = blockIdx.x * blockDim.x + threadIdx.x;
    if (i < n) {
        float d = deg[i];
        deg[i] = (d > 0.0f) ? rsqrtf(d) : 0.0f;
    }
}

__global__ void zero_f32_kernel(float* p, int n) {
    int i = blockIdx.x * blockDim.x + threadIdx.x;
    if (i < n) p[i] = 0.0f;
}

// ---------------------------------------------------------------------------
// fp32 WMMA GEMM (compile-time K, Nc):  C[M x Nc] = A[M x K] @ B[K x Nc] (+bias)
// One block = 16 rows of A staged in LDS (float4 fill, pitch K+4 so column
// fragment reads hit distinct banks), one wave per 16-col output tile.
// All B fragments are preloaded to registers before the WMMA loop, so the
// steady state is pure ds_load + v_wmma. blockDim.x = 32*(Nc/16).
// ---------------------------------------------------------------------------
template <int K, int Nc>
__global__ void gemm_bias_wmma(const float* __restrict__ A,
                               const float* __restrict__ Bm,
                               const float* __restrict__ bias,
                               float* __restrict__ C, int M) {
    constexpr int PITCH = K + 4;                  // float4-aligned, bank-spread
    __shared__ float lds[16 * PITCH];
    const int tid  = threadIdx.x;
    const int lane = tid & 31;
    const int wave = tid >> 5;
    const int row0 = blockIdx.x * 16;

    const int m    = lane & 15;          // M (and N) index held by this lane
    const int kb   = (lane >> 4) << 1;   // lanes 16-31 hold K+2..K+3
    const int ncol = (wave << 4) + m;    // output column

    // preload all B fragments for this wave's column (loop-invariant weights)
    v2f bfrag[K / 4];
#pragma unroll
    for (int kk = 0; kk < K / 4; ++kk) {
        bfrag[kk].x = Bm[(size_t)(kk * 4 + kb) * Nc + ncol];
        bfrag[kk].y = Bm[(size_t)(kk * 4 + kb + 1) * Nc + ncol];
    }

    // cooperative float4 fill of the 16 x K A-tile
    constexpr int NV4 = 16 * K / 4;
    for (int idx = tid; idx < NV4; idx += blockDim.x) {
        int r  = idx / (K / 4);
        int c4 = idx - r * (K / 4);
        float4 v = reinterpret_cast<const float4*>(A)[(size_t)(row0 + r) * (K / 4) + c4];
        *reinterpret_cast<float4*>(&lds[r * PITCH + c4 * 4]) = v;
    }
    __syncthreads();

    v8f c = {0.f, 0.f, 0.f, 0.f, 0.f, 0.f, 0.f, 0.f};
#pragma unroll
    for (int kk = 0; kk < K / 4; ++kk) {
        v2f a;
        a.x = lds[m * PITCH + kk * 4 + kb];
        a.y = lds[m * PITCH + kk * 4 + kb + 1];
        c = __builtin_amdgcn_wmma_f32_16x16x4_f32(
                /*neg_a=*/false, a, /*neg_b=*/false, bfrag[kk],
                /*c_mod=*/(short)0, c, /*reuse_a=*/false, /*reuse_b=*/false);
    }

    // C/D layout: VGPR j -> M = j (lanes 0-15), M = 8+j (lanes 16-31)
    const int m8 = (lane >> 4) << 3;
#pragma unroll
    for (int j = 0; j < 8; ++j) {
        int row = row0 + m8 + j;
        float v = c[j];
        if (bias) v += bias[ncol];
        C[(size_t)row * Nc + ncol] = v;
    }
}

// ---------------------------------------------------------------------------
// Edge scatter: agg[b, col, h] += hw[b, row, h] * dinv[row]*dinv[col]
// one thread per (edge, h), loops over B=4 batches. Atomic f32 adds go to L2.
// ---------------------------------------------------------------------------
__global__ void scatter_edges_kernel(const float* __restrict__ HW,
                                     const long long* __restrict__ ei,
                                     const float* __restrict__ dinv,
                                     float* __restrict__ AGG,
                                     int Nn, int E) {
    long long idx = (long long)blockIdx.x * blockDim.x + threadIdx.x;
    int e = (int)(idx >> 6);
    int h = (int)(idx & 63);
    if (e >= E) return;
    int r = (int)ei[e];
    int c = (int)ei[(size_t)E + e];
    float w = dinv[r] * dinv[c];
#pragma unroll
    for (int b = 0; b < 4; ++b) {
        size_t ro = ((size_t)b * Nn + r) * 64 + h;
        size_t co = ((size_t)b * Nn + c) * 64 + h;
        atomicAdd(&AGG[co], HW[ro] * w);
    }
}

// self-loop term + bias + ReLU, fused (kernels are stream-serialized, so a
// plain read-modify-write is safe here)
__global__ void finish_gcn_kernel(const float* __restrict__ HW,
                                  const float* __restrict__ dinv,
                                  const float* __restrict__ bias,
                                  float* __restrict__ AGG,
                                  int Nn, int total) {
    int idx = blockIdx.x * blockDim.x + threadIdx.x;
    if (idx >= total) return;
    int h = idx & 63;
    int n = (idx >> 6) % Nn;
    float d = dinv[n];
    float v = AGG[idx] + HW[idx] * d * d + bias[h];
    AGG[idx] = v > 0.0f ? v : 0.0f;
}

// ---------------------------------------------------------------------------
// Temporal conv (k=3 over node axis) as 3 accumulated WMMA GEMM taps.
// Block: 4 waves, 16 output rows of one batch; LDS holds 18x64 halo tile
// (float4 fill, pitch 68). Per-tap weight fragments preloaded to registers.
// y[b,n,cout] = relu( sum_{t,cin} h[b, n+t-1, cin] * Wt[cout,cin,t] + bt )
// ---------------------------------------------------------------------------
__global__ void tconv_wmma(const float* __restrict__ H2,
                           const float* __restrict__ Wt,
                           const float* __restrict__ bt,
                           float* __restrict__ Y, int Nn) {
    constexpr int PITCH = 68;
    __shared__ float tile[18 * PITCH];
    const int tid  = threadIdx.x;
    const int lane = tid & 31;
    const int wave = tid >> 5;  // 0..3
    const int n0 = blockIdx.x * 16;
    const int b  = blockIdx.y;
    const size_t base = (size_t)b * Nn * 64;

    // float4 fill of the 18-row halo tile (zero-padded at batch edges)
    for (int idx = tid; idx < 18 * 16; idx += blockDim.x) {
        int r  = idx >> 4;        // 0..17
        int c4 = idx & 15;        // float4 column
        int n = n0 - 1 + r;
        float4 v = make_float4(0.f, 0.f, 0.f, 0.f);
        if (n >= 0 && n < Nn)
            v = reinterpret_cast<const float4*>(H2)[(base >> 2) + (size_t)n * 16 + c4];
        *reinterpret_cast<float4*>(&tile[r * PITCH + c4 * 4]) = v;
    }
    __syncthreads();

    const int m    = lane & 15;
    const int kb   = (lane >> 4) << 1;
    const int cout = (wave << 4) + m;

    v8f c = {0.f, 0.f, 0.f, 0.f, 0.f, 0.f, 0.f, 0.f};
#pragma unroll
    for (int t = 0; t < 3; ++t) {
        // preload this tap's weight fragments: B[K=cin, N=cout] = Wt[cout,cin,t]
        v2f wfrag[16];
#pragma unroll
        for (int kk = 0; kk < 16; ++kk) {
            wfrag[kk].x = Wt[((size_t)cout * 64 + kk * 4 + kb) * 3 + t];
            wfrag[kk].y = Wt[((size_t)cout * 64 + kk * 4 + kb + 1) * 3 + t];
        }
#pragma unroll
        for (int kk = 0; kk < 16; ++kk) {
            v2f a;
            a.x = tile[(m + t) * PITCH + kk * 4 + kb];   // input row = (m+1)+(t-1)
            a.y = tile[(m + t) * PITCH + kk * 4 + kb + 1];
            c = __builtin_amdgcn_wmma_f32_16x16x4_f32(
                    false, a, false, wfrag[kk], (short)0, c, false, false);
        }
    }

    const int m8 = (lane >> 4) << 3;
#pragma unroll
    for (int j = 0; j < 8; ++j) {
        int n = n0 + m8 + j;
        float v = c[j] + bt[cout];
        v = v > 0.0f ? v : 0.0f;
        Y[base + (size_t)n * 64 + cout] = v;
    }
}

// ---------------------------------------------------------------------------
extern "C" void kernel_launch(void* const* d_in, const int* in_sizes, int n_in,
                              void* d_out, int out_size, void* d_ws, size_t ws_size,
                              hipStream_t stream) {
    const float*     x  = (const float*)d_in[0];
    const long long* ei = (const long long*)d_in[1];
    const float*     W1 = (const float*)d_in[2];
    const float*     b1 = (const float*)d_in[3];
    const float*     W2 = (const float*)d_in[4];
    const float*     b2 = (const float*)d_in[5];
    const float*     Wt = (const float*)d_in[6];
    const float*     bt = (const float*)d_in[7];
    const float*     Wf = (const float*)d_in[8];
    const float*     bf = (const float*)d_in[9];
    float* out = (float*)d_out;

    const int Bc  = 4;
    const int Fin = 128;
    const int E   = in_sizes[1] / 2;
    const int Nn  = in_sizes[0] / (Bc * Fin);  // 20000
    const int M   = Bc * Nn;                   // 80000 rows
    const int tot = M * 64;

    float* ws   = (float*)d_ws;
    float* dinv = ws;                               // Nn floats (deg -> dinv)
    float* bufA = ws + 20480;                       // M*64 floats
    float* bufB = bufA + (size_t)M * 64;            // M*64 floats

    const int T = 256;

    // degree + normalization
    init_deg_kernel<<<(Nn + T - 1) / T, T, 0, stream>>>(dinv, Nn);
    accum_deg_kernel<<<(E + T - 1) / T, T, 0, stream>>>(ei, dinv, E);
    dinv_kernel<<<(Nn + T - 1) / T, T, 0, stream>>>(dinv, Nn);

    // ---- GCN layer 1 ----
    gemm_bias_wmma<128, 64><<<M / 16, 128, 0, stream>>>(x, W1, nullptr, bufA, M);
    zero_f32_kernel<<<(tot + T - 1) / T, T, 0, stream>>>(bufB, tot);
    {
        long long threads = (long long)E * 64;
        int blocks = (int)((threads + T - 1) / T);
        scatter_edges_kernel<<<blocks, T, 0, stream>>>(bufA, ei, dinv, bufB, Nn, E);
    }
    finish_gcn_kernel<<<(tot + T - 1) / T, T, 0, stream>>>(bufA, dinv, b1, bufB, Nn, tot);

    // ---- GCN layer 2 ----
    gemm_bias_wmma<64, 64><<<M / 16, 128, 0, stream>>>(bufB, W2, nullptr, bufA, M);
    zero_f32_kernel<<<(tot + T - 1) / T, T, 0, stream>>>(bufB, tot);
    {
        long long threads = (long long)E * 64;
        int blocks = (int)((threads + T - 1) / T);
        scatter_edges_kernel<<<blocks, T, 0, stream>>>(bufA, ei, dinv, bufB, Nn, E);
    }
    finish_gcn_kernel<<<(tot + T - 1) / T, T, 0, stream>>>(bufA, dinv, b2, bufB, Nn, tot);

    // ---- temporal conv (bias+relu fused) ----
    tconv_wmma<<<dim3(Nn / 16, Bc), 128, 0, stream>>>(bufB, Wt, bt, bufA, Nn);

    // ---- final projection (bias fused) -> d_out ----
    gemm_bias_wmma<64, 128><<<M / 16, 256, 0, stream>>>(bufA, Wf, bf, out, M);
}